// MVFFeatureNetDVP_22625887715564
// MI455X (gfx1250) — compile-verified
//
#include <hip/hip_runtime.h>
#include <hip/hip_bf16.h>
#include <float.h>

typedef __attribute__((ext_vector_type(2))) float v2f;
typedef __attribute__((ext_vector_type(8))) float v8f;

#define BEV_HW 512

// ---------------------------------------------------------------- fill kernels
__global__ void fill_f32_kernel(float* __restrict__ p, float v, long long n) {
    long long i = (long long)blockIdx.x * blockDim.x + threadIdx.x;
    if (i < n) p[i] = v;
}
__global__ void fill_i32_kernel(int* __restrict__ p, int v, long long n) {
    long long i = (long long)blockIdx.x * blockDim.x + threadIdx.x;
    if (i < n) p[i] = v;
}

// ------------------------------------------------- pass 1: per-voxel count/sum
__global__ __launch_bounds__(256) void voxel_accum_kernel(
    const float* __restrict__ coord, const int* __restrict__ pv,
    float* __restrict__ counts, float* __restrict__ sums, int Npts) {
    int p = blockIdx.x * blockDim.x + threadIdx.x;
    if (p >= Npts) return;
    int v = pv[p];
    atomicAdd(&counts[v], 1.0f);
    atomicAdd(&sums[3 * v + 0], coord[3 * p + 0]);
    atomicAdd(&sums[3 * v + 1], coord[3 * p + 1]);
    atomicAdd(&sums[3 * v + 2], coord[3 * p + 2]);
}

// ------------------------------------------ pass 2: 65 BN sufficient statistics
// gstats layout: [0..54] = sym outer product Sum f_i f_j (i<=j), [55..64] = Sum f
__global__ __launch_bounds__(256) void stats_kernel(
    const float* __restrict__ coord, const float* __restrict__ lcoord,
    const float* __restrict__ inten, const int* __restrict__ pv,
    const float* __restrict__ counts, const float* __restrict__ sums,
    float* __restrict__ gstats, int Npts) {
    __shared__ float s[65];
    int t = threadIdx.x;
    if (t < 65) s[t] = 0.0f;
    __syncthreads();

    int p = blockIdx.x * blockDim.x + t;
    float f[10];
#pragma unroll
    for (int k = 0; k < 10; ++k) f[k] = 0.0f;
    if (p < Npts) {
        float c0 = coord[3 * p + 0], c1 = coord[3 * p + 1], c2 = coord[3 * p + 2];
        int v = pv[p];
        float cnt = counts[v];
        cnt = cnt < 1.0f ? 1.0f : cnt;
        float inv = 1.0f / cnt;
        f[0] = c0; f[1] = c1; f[2] = c2;
        f[3] = inten[p];
        f[4] = c0 - sums[3 * v + 0] * inv;
        f[5] = c1 - sums[3 * v + 1] * inv;
        f[6] = c2 - sums[3 * v + 2] * inv;
        f[7] = lcoord[3 * p + 0]; f[8] = lcoord[3 * p + 1]; f[9] = lcoord[3 * p + 2];
    }
    float acc[65];
    int idx = 0;
#pragma unroll
    for (int i = 0; i < 10; ++i)
#pragma unroll
        for (int j = i; j < 10; ++j) acc[idx++] = f[i] * f[j];
#pragma unroll
    for (int k = 0; k < 10; ++k) acc[55 + k] = f[k];

#pragma unroll
    for (int m = 0; m < 65; ++m) {
        float v = acc[m];
        for (int o = 16; o > 0; o >>= 1) v += __shfl_down(v, o, 32);
        if ((threadIdx.x & 31) == 0) atomicAdd(&s[m], v);
    }
    __syncthreads();
    if (t < 65) atomicAdd(&gstats[t], s[t]);
}

// ------------------------------------------------ BN finalize: scale/bias[64]
__global__ void bn_finalize_kernel(
    const float* __restrict__ gstats, const float* __restrict__ Wm,
    const float* __restrict__ gamma, const float* __restrict__ beta,
    float* __restrict__ scale, float* __restrict__ bias, float invN) {
    int c = threadIdx.x;  // 64 threads
    float w[10];
#pragma unroll
    for (int k = 0; k < 10; ++k) w[k] = Wm[c * 10 + k];
    float mu = 0.0f;
#pragma unroll
    for (int k = 0; k < 10; ++k) mu += w[k] * gstats[55 + k];
    mu *= invN;
    float ex2 = 0.0f;
    int idx = 0;
#pragma unroll
    for (int i = 0; i < 10; ++i)
#pragma unroll
        for (int j = i; j < 10; ++j) {
            float sij = gstats[idx++];
            ex2 += (i == j ? w[i] * w[i] : 2.0f * w[i] * w[j]) * sij;
        }
    ex2 *= invN;
    float var = ex2 - mu * mu;
    var = var < 0.0f ? 0.0f : var;
    float sc = gamma[c] * rsqrtf(var + 1e-3f);
    scale[c] = sc;
    bias[c] = beta[c] - mu * sc;
}

// ---------------- fused: feature build + WMMA GEMM + BN affine + ReLU + scatter-max
// One wave handles a 16-point tile, all 64 output channels, via
// V_WMMA_F32_16X16X4_F32: 3 K-chunks (K=10 padded to 12) x 4 N-tiles of 16 ch.
// Fragments are built from named scalars with single hi/lo selects: no dynamic
// register-array indexing (avoids v_cmp/v_cndmask select trees).
__global__ __launch_bounds__(256) void gemm_scatter_kernel(
    const float* __restrict__ coord, const float* __restrict__ lcoord,
    const float* __restrict__ inten, const int* __restrict__ pv,
    const float* __restrict__ Wm, const float* __restrict__ counts,
    const float* __restrict__ sums, const float* __restrict__ scale,
    const float* __restrict__ bias, int* __restrict__ vmax, int Npts) {
    int wave = (blockIdx.x * blockDim.x + threadIdx.x) >> 5;  // uniform per wave
    int lane = threadIdx.x & 31;
    int tb = wave * 16;
    if (tb >= Npts) return;  // wave-uniform exit; EXEC stays full otherwise

    int l16 = lane & 15;
    bool hi = lane >= 16;
    int lastp = Npts - 1;

    // A-fragment source point: lanes 0-15 hold rows M=0..15 (ISA 16x4 f32 layout)
    float f0 = 0.f, f1 = 0.f, f2 = 0.f, f3 = 0.f, f4 = 0.f;
    float f5 = 0.f, f6 = 0.f, f7 = 0.f, f8 = 0.f, f9 = 0.f;
    {
        int pA = tb + l16;
        int pc = pA < Npts ? pA : lastp;        // clamped, always-valid address
        float c0 = coord[3 * pc + 0], c1 = coord[3 * pc + 1], c2 = coord[3 * pc + 2];
        int v = pv[pc];
        float cnt = counts[v];
        cnt = cnt < 1.0f ? 1.0f : cnt;
        float inv = 1.0f / cnt;
        float itv = inten[pc];
        float l0 = lcoord[3 * pc + 0], l1 = lcoord[3 * pc + 1], l2 = lcoord[3 * pc + 2];
        float m0 = c0 - sums[3 * v + 0] * inv;
        float m1 = c1 - sums[3 * v + 1] * inv;
        float m2 = c2 - sums[3 * v + 2] * inv;
        if (pA < Npts) {
            f0 = c0; f1 = c1; f2 = c2; f3 = itv;
            f4 = m0; f5 = m1; f6 = m2;
            f7 = l0; f8 = l1; f9 = l2;
        }
    }

    // A fragments (16x4 f32 layout): lanes<16 carry K={4kc,4kc+1}, lanes>=16
    // carry K={4kc+2,4kc+3}; K=10,11 are zero padding.
    v2f a0, a1, a2;
    a0.x = hi ? f2 : f0;   a0.y = hi ? f3 : f1;
    a1.x = hi ? f6 : f4;   a1.y = hi ? f7 : f5;
    a2.x = hi ? 0.f : f8;  a2.y = hi ? 0.f : f9;

    // B fragments: B[k][n] = W[n][k], same K split across lane halves.
    int ko = hi ? 2 : 0;
    v2f b0[4], b1[4], b2[4];
#pragma unroll
    for (int nt = 0; nt < 4; ++nt) {
        const float* wr = Wm + (nt * 16 + l16) * 10;
        b0[nt].x = wr[ko];      b0[nt].y = wr[ko + 1];
        b1[nt].x = wr[ko + 4];  b1[nt].y = wr[ko + 5];
        float w8 = wr[8], w9 = wr[9];
        b2[nt].x = hi ? 0.f : w8;
        b2[nt].y = hi ? 0.f : w9;
    }

    v8f acc[4];
#pragma unroll
    for (int nt = 0; nt < 4; ++nt) acc[nt] = (v8f){0, 0, 0, 0, 0, 0, 0, 0};
#pragma unroll
    for (int nt = 0; nt < 4; ++nt)
        acc[nt] = __builtin_amdgcn_wmma_f32_16x16x4_f32(
            false, a0, false, b0[nt], (short)0, acc[nt], false, false);
#pragma unroll
    for (int nt = 0; nt < 4; ++nt)
        acc[nt] = __builtin_amdgcn_wmma_f32_16x16x4_f32(
            false, a1, false, b1[nt], (short)0, acc[nt], false, false);
#pragma unroll
    for (int nt = 0; nt < 4; ++nt)
        acc[nt] = __builtin_amdgcn_wmma_f32_16x16x4_f32(
            false, a2, false, b2[nt], (short)0, acc[nt], false, false);

    // C/D layout: VGPR r, lanes 0-15 -> row r, lanes 16-31 -> row r+8
    int moff = hi ? 8 : 0;
    int pvs[8];
#pragma unroll
    for (int r = 0; r < 8; ++r) {
        int p = tb + moff + r;
        int pc = p < Npts ? p : lastp;          // unconditional valid load
        int vv = pv[pc];
        pvs[r] = (p < Npts) ? vv : -1;
    }
#pragma unroll
    for (int nt = 0; nt < 4; ++nt) {
        int ch = nt * 16 + l16;
        float sc = scale[ch], bi = bias[ch];
#pragma unroll
        for (int r = 0; r < 8; ++r) {
            if (pvs[r] >= 0) {
                float val = acc[nt][r] * sc + bi;
                val = val > 0.0f ? val : 0.0f;
                // post-ReLU values >= 0: int-compare == float-compare; sentinel
                // bits(-FLT_MAX) is a negative int, so any point beats it.
                atomicMax(&vmax[(long long)pvs[r] * 64 + ch], __float_as_int(val));
            }
        }
    }
}

// ------------------------------------------- dense scatter: [B,64,H,W] output
__global__ __launch_bounds__(256) void scatter_dense_kernel(
    const int* __restrict__ vmax, const int* __restrict__ vf,
    float* __restrict__ out, int V) {
    long long i = (long long)blockIdx.x * blockDim.x + threadIdx.x;
    if (i >= 64LL * V) return;
    int c = (int)(i / V);      // consecutive lanes -> consecutive voxels
    int v = (int)(i - (long long)c * V);
    int bb = vf[3 * v + 0], h = vf[3 * v + 1], w = vf[3 * v + 2];
    float val = __int_as_float(vmax[(long long)v * 64 + c]);
    out[(((long long)bb * 64 + c) * BEV_HW + h) * BEV_HW + w] = val;
}

// ---------------------------------------------------------------------- launch
extern "C" void kernel_launch(void* const* d_in, const int* in_sizes, int n_in,
                              void* d_out, int out_size, void* d_ws, size_t ws_size,
                              hipStream_t stream) {
    const float* coord  = (const float*)d_in[0];
    const float* lcoord = (const float*)d_in[1];
    const float* inten  = (const float*)d_in[2];
    const int*   pv     = (const int*)d_in[3];
    const int*   vf     = (const int*)d_in[4];
    const float* Wm     = (const float*)d_in[5];
    const float* gamma  = (const float*)d_in[6];
    const float* beta   = (const float*)d_in[7];
    int N = in_sizes[2];
    int V = in_sizes[4] / 3;

    // workspace layout (floats): counts[V] | sums[3V] | gstats[65..pad 128] |
    //                            scale[64] | bias[64] | (align) vmax[64V] ints
    float* w      = (float*)d_ws;
    float* counts = w;
    float* sums   = w + V;
    float* gstats = w + 4LL * V;
    float* scale  = gstats + 128;
    float* bias   = scale + 64;
    int*   vmax   = (int*)(w + 4LL * V + 256);
    float* out    = (float*)d_out;

    // bit pattern of -FLT_MAX (host-side constant; __float_as_int is device-only)
    const int NEG_FLT_MAX_BITS = (int)0xFF7FFFFFu;

    const int T = 256;
    long long nInit = 4LL * V + 65;           // counts+sums+gstats contiguous
    fill_f32_kernel<<<(int)((nInit + T - 1) / T), T, 0, stream>>>(counts, 0.0f, nInit);
    fill_i32_kernel<<<(int)((64LL * V + T - 1) / T), T, 0, stream>>>(
        vmax, NEG_FLT_MAX_BITS, 64LL * V);
    fill_f32_kernel<<<(int)(((long long)out_size + T - 1) / T), T, 0, stream>>>(
        out, 0.0f, (long long)out_size);

    voxel_accum_kernel<<<(N + T - 1) / T, T, 0, stream>>>(coord, pv, counts, sums, N);

    stats_kernel<<<(N + T - 1) / T, T, 0, stream>>>(
        coord, lcoord, inten, pv, counts, sums, gstats, N);

    bn_finalize_kernel<<<1, 64, 0, stream>>>(gstats, Wm, gamma, beta, scale, bias,
                                             1.0f / (float)N);

    long long nWaves = ((long long)N + 15) / 16;
    long long nThr   = nWaves * 32;
    gemm_scatter_kernel<<<(int)((nThr + T - 1) / T), T, 0, stream>>>(
        coord, lcoord, inten, pv, Wm, counts, sums, scale, bias, vmax, N);

    scatter_dense_kernel<<<(int)((64LL * V + T - 1) / T), T, 0, stream>>>(
        vmax, vf, out, V);
}